// MatNet_88940182765824
// MI455X (gfx1250) — compile-verified
//
#include <hip/hip_runtime.h>
#include <math.h>

typedef __attribute__((ext_vector_type(2))) float v2f;
typedef __attribute__((ext_vector_type(8))) float v8f;
typedef int v4i_t __attribute__((vector_size(16)));
typedef __attribute__((address_space(1))) v4i_t* v4i_gas;   // global int4*
typedef __attribute__((address_space(3))) v4i_t* v4i_las;   // LDS int4*

#define TILE   64
#define TPB    128   // 4 waves (wave32)
#define TPC    8     // column tiles per chunk
#define NDIM   6000
#define STRIPS ((NDIM + TILE - 1) / TILE)   // 94
#define CHUNKS ((STRIPS + TPC - 1) / TPC)   // 12

#if defined(__has_builtin)
#if __has_builtin(__builtin_amdgcn_global_load_async_to_lds_b128) && \
    __has_builtin(__builtin_amdgcn_s_wait_asynccnt)
#define USE_ASYNC_LDS 1
#endif
#endif
#ifndef USE_ASYNC_LDS
#define USE_ASYNC_LDS 0
#endif

__device__ __forceinline__ void atomAddF(float* p, float v) {
  __hip_atomic_fetch_add(p, v, __ATOMIC_RELAXED, __HIP_MEMORY_SCOPE_AGENT);
}

__global__ void zero_f32(float* p, int n) {
  int i = blockIdx.x * blockDim.x + threadIdx.x;
  if (i < n) p[i] = 0.f;
}

// out[i] = sum_k part[k*len + i]
__global__ void reduce_sum(const float* __restrict__ part, float* __restrict__ out,
                           int K, int len) {
  int i = blockIdx.x * blockDim.x + threadIdx.x;
  if (i >= len) return;
  float s = 0.f;
  for (int k = 0; k < K; ++k) s += part[(size_t)k * (size_t)len + i];
  out[i] = s;
}

// ---------------------------------------------------------------------------
// Fused strip pass over Y0. Block = (column chunk, 64-row strip).
//   E = clamp(Y0 - (useUV ? Wl*Wr^T : 0), +-ths[tidx])
//   partR[chunk][N x 5]  = E   @ Wr   over this block's K-span (reg-accumulated)
//   partL[strip][M x 5]  = E^T @ Wl   per column tile (disjoint plain stores)
// Y0 tiles staged to LDS with GLOBAL_LOAD_ASYNC_TO_LDS_B128, double buffered,
// overlapped with V_WMMA_F32_16X16X4_F32 chains. No global atomics.
// ---------------------------------------------------------------------------
__global__ __launch_bounds__(TPB)
void fused_pass(const float* __restrict__ Y0,
                const float* __restrict__ ths, int tidx,
                const float* __restrict__ Wl,   // N x 5 (row side) or null
                const float* __restrict__ Wr,   // M x 5 (col side) or null
                float* __restrict__ partR,      // [CHUNKS][n*5] or null
                float* __restrict__ partL,      // [STRIPS][m*5] or null
                int useUV, int n, int m, int tpc)
{
#if USE_ASYNC_LDS
  __shared__ float rawT[2][TILE][TILE];
#endif
  __shared__ float Et[TILE][TILE + 1];
  __shared__ float WlT[TILE][16];      // cols 5..15 zero (B-fragment padding)
  __shared__ float WrT[2][TILE][16];

  const int tid  = threadIdx.x;
  const int rb   = blockIdx.y * TILE;
  const int Tt   = (m + TILE - 1) / TILE;
  const int t0   = blockIdx.x * tpc;
  const int t1   = (t0 + tpc < Tt) ? (t0 + tpc) : Tt;
  const float th = ths[tidx];

  const int w    = tid >> 5;   // wave 0..3
  const int lane = tid & 31;
  const int m16  = lane & 15;
  const int half = lane >> 4;

  // row-side panel: fixed for the whole strip
  for (int idx = tid; idx < TILE * 16; idx += TPB) {
    int r = idx >> 4, c = idx & 15;
    float vl = 0.f;
    if (c < 5 && Wl && rb + r < n) vl = Wl[(rb + r) * 5 + c];
    WlT[r][c] = vl;
  }

#if USE_ASYNC_LDS
#define ISSUE_TILE(TI, BUF)                                                    \
  do {                                                                         \
    int _cb = (TI) * TILE;                                                     \
    _Pragma("unroll")                                                          \
    for (int _k = 0; _k < 8; ++_k) {                                           \
      int _q = tid + _k * TPB;                                                 \
      int _r = _q >> 4, _jg = _q & 15;                                         \
      int _gr = rb + _r;       _gr = (_gr < n) ? _gr : (n - 1);                \
      int _gc = _cb + _jg * 4; _gc = (_gc + 4 <= m) ? _gc : (m - 4);           \
      const float* _gp = Y0 + (size_t)_gr * (size_t)m + _gc;                   \
      float* _lp = &rawT[BUF][_r][_jg * 4];                                    \
      __builtin_amdgcn_global_load_async_to_lds_b128(                          \
          (v4i_gas)_gp, (v4i_las)_lp, 0, 0);                                   \
    }                                                                          \
  } while (0)
  if (t0 < t1) ISSUE_TILE(t0, t0 & 1);
#endif

  v8f accR = {0.f, 0.f, 0.f, 0.f, 0.f, 0.f, 0.f, 0.f};

  for (int ti = t0; ti < t1; ++ti) {
    const int buf = ti & 1;
    const int cb  = ti * TILE;

#if USE_ASYNC_LDS
    if (ti + 1 < t1) {
      ISSUE_TILE(ti + 1, (ti + 1) & 1);
      __builtin_amdgcn_s_wait_asynccnt(8);   // previous tile landed; next in flight
    } else {
      __builtin_amdgcn_s_wait_asynccnt(0);
    }
#endif
    // column-side panel for this tile (double buffered)
    for (int idx = tid; idx < TILE * 16; idx += TPB) {
      int r = idx >> 4, c = idx & 15;
      float vr = 0.f;
      if (c < 5 && Wr && cb + r < m) vr = Wr[(cb + r) * 5 + c];
      WrT[buf][r][c] = vr;
    }
    __syncthreads();   // raw tile + panels visible

    // build E in LDS
    for (int idx = tid; idx < TILE * TILE; idx += TPB) {
      int i = idx >> 6, j = idx & 63;
      int gi = rb + i, gj = cb + j;
      float e = 0.f;
      if (gi < n && gj < m) {
#if USE_ASYNC_LDS
        float y = rawT[buf][i][j];
#else
        float y = Y0[(size_t)gi * (size_t)m + gj];
#endif
        if (useUV) {
          float d = 0.f;
          #pragma unroll
          for (int k = 0; k < 5; ++k) d += WlT[i][k] * WrT[buf][j][k];
          y -= d;
        }
        e = fminf(fmaxf(y, -th), th);   // x - thre(x,th) == clamp(x,+-th)
      }
      Et[i][j] = e;
    }
    __syncthreads();

    if (partR) {
      // accR (16x16, 5 valid cols) += E[strip rows] @ Wr, K accumulated across chunk
      #pragma unroll
      for (int kk = 0; kk < TILE; kk += 4) {
        int k0 = kk + 2 * half;
        v2f a, b;
        a.x = Et[16 * w + m16][k0];
        a.y = Et[16 * w + m16][k0 + 1];
        b.x = WrT[buf][k0][m16];
        b.y = WrT[buf][k0 + 1][m16];
        accR = __builtin_amdgcn_wmma_f32_16x16x4_f32(false, a, false, b,
                                                     (short)0, accR, false, false);
      }
    }

    if (partL) {
      // accL (16x16) = E^T[tile cols] @ Wl over this strip's 64 rows
      v8f accL = {0.f, 0.f, 0.f, 0.f, 0.f, 0.f, 0.f, 0.f};
      #pragma unroll
      for (int kk = 0; kk < TILE; kk += 4) {
        int k0 = kk + 2 * half;
        v2f a, b;
        a.x = Et[k0][16 * w + m16];     // transposed read of E
        a.y = Et[k0 + 1][16 * w + m16];
        b.x = WlT[k0][m16];
        b.y = WlT[k0 + 1][m16];
        accL = __builtin_amdgcn_wmma_f32_16x16x4_f32(false, a, false, b,
                                                     (short)0, accL, false, false);
      }
      if (m16 < 5) {
        float* dst = partL + (size_t)blockIdx.y * (size_t)(m * 5);
        #pragma unroll
        for (int v = 0; v < 8; ++v) {
          int col = cb + 16 * w + v + 8 * half;
          if (col < m) dst[col * 5 + m16] = accL[v];   // disjoint plain store
        }
      }
    }
    __syncthreads();   // Et / panels free for next tile
  }

  if (partR && m16 < 5) {
    float* dst = partR + (size_t)blockIdx.x * (size_t)(n * 5);
    #pragma unroll
    for (int v = 0; v < 8; ++v) {
      int row = rb + 16 * w + v + 8 * half;
      if (row < n) dst[row * 5 + m16] = accR[v];       // disjoint plain store
    }
  }
}

// ---------------------------------------------------------------------------
// G(5x5) += X^T Y over n rows (X,Y are n x 5). G pre-zeroed.
// ---------------------------------------------------------------------------
__global__ void gram5(const float* __restrict__ X, const float* __restrict__ Y,
                      float* __restrict__ G, int n)
{
  __shared__ float red[256];
  float acc[25];
  #pragma unroll
  for (int e = 0; e < 25; ++e) acc[e] = 0.f;
  for (int i = blockIdx.x * blockDim.x + threadIdx.x; i < n;
       i += gridDim.x * blockDim.x) {
    float x[5], y[5];
    #pragma unroll
    for (int c = 0; c < 5; ++c) { x[c] = X[i * 5 + c]; y[c] = Y[i * 5 + c]; }
    #pragma unroll
    for (int a = 0; a < 5; ++a)
      #pragma unroll
      for (int b = 0; b < 5; ++b) acc[a * 5 + b] += x[a] * y[b];
  }
  for (int e = 0; e < 25; ++e) {
    red[threadIdx.x] = acc[e];
    __syncthreads();
    for (int s = blockDim.x >> 1; s > 0; s >>= 1) {
      if (threadIdx.x < (unsigned)s) red[threadIdx.x] += red[threadIdx.x + s];
      __syncthreads();
    }
    if (threadIdx.x == 0) atomAddF(&G[e], red[0]);
    __syncthreads();
  }
}

// out[n x 5] = beta * X + alpha * (Y @ M5);  alpha = alpha_arr? alpha_arr[aidx] : 1
__global__ void rightmul5(float* __restrict__ out, const float* __restrict__ X,
                          const float* __restrict__ Y, const float* __restrict__ M5,
                          const float* __restrict__ alpha_arr, int aidx,
                          float beta, int n)
{
  int i = blockIdx.x * blockDim.x + threadIdx.x;
  if (i >= n) return;
  float alpha = alpha_arr ? alpha_arr[aidx] : 1.0f;
  float y[5];
  #pragma unroll
  for (int c = 0; c < 5; ++c) y[c] = Y[i * 5 + c];
  #pragma unroll
  for (int c = 0; c < 5; ++c) {
    float s = 0.f;
    #pragma unroll
    for (int k = 0; k < 5; ++k) s += y[k] * M5[k * 5 + c];
    float base = X ? beta * X[i * 5 + c] : 0.f;
    out[i * 5 + c] = base + alpha * s;
  }
}

// CholeskyQR helper: G = R^T R, output Rinv (upper) so Q = X @ Rinv
__global__ void chol_rinv5(const float* __restrict__ G, float* __restrict__ Rinv)
{
  float R[5][5], Ri[5][5];
  for (int i = 0; i < 5; ++i)
    for (int j = 0; j < 5; ++j) { R[i][j] = 0.f; Ri[i][j] = 0.f; }
  for (int j = 0; j < 5; ++j) {
    float s = G[j * 5 + j];
    for (int k = 0; k < j; ++k) s -= R[k][j] * R[k][j];
    R[j][j] = sqrtf(fmaxf(s, 1e-30f));
    for (int i = j + 1; i < 5; ++i) {
      float t = G[j * 5 + i];
      for (int k = 0; k < j; ++k) t -= R[k][j] * R[k][i];
      R[j][i] = t / R[j][j];
    }
  }
  for (int j = 0; j < 5; ++j) {
    Ri[j][j] = 1.f / R[j][j];
    for (int i = j - 1; i >= 0; --i) {
      float s = 0.f;
      for (int k = i + 1; k <= j; ++k) s += R[i][k] * Ri[k][j];
      Ri[i][j] = -s / R[i][i];
    }
  }
  for (int i = 0; i < 5; ++i)
    for (int j = 0; j < 5; ++j) Rinv[i * 5 + j] = Ri[i][j];
}

// 5x5 inverse, Gauss-Jordan with partial pivoting
__global__ void inv5(const float* __restrict__ A, float* __restrict__ Ainv)
{
  float aug[5][10];
  for (int i = 0; i < 5; ++i)
    for (int j = 0; j < 5; ++j) { aug[i][j] = A[i * 5 + j]; aug[i][j + 5] = (i == j) ? 1.f : 0.f; }
  for (int c = 0; c < 5; ++c) {
    int piv = c;
    for (int r = c + 1; r < 5; ++r)
      if (fabsf(aug[r][c]) > fabsf(aug[piv][c])) piv = r;
    if (piv != c)
      for (int j = 0; j < 10; ++j) { float t = aug[c][j]; aug[c][j] = aug[piv][j]; aug[piv][j] = t; }
    float p = aug[c][c];
    p = (fabsf(p) > 1e-30f) ? p : 1e-30f;
    float ip = 1.f / p;
    for (int j = 0; j < 10; ++j) aug[c][j] *= ip;
    for (int r = 0; r < 5; ++r) {
      if (r == c) continue;
      float f = aug[r][c];
      for (int j = 0; j < 10; ++j) aug[r][j] -= f * aug[c][j];
    }
  }
  for (int i = 0; i < 5; ++i)
    for (int j = 0; j < 5; ++j) Ainv[i * 5 + j] = aug[i][j + 5];
}

// 5x5 symmetric eigendecomposition (cyclic Jacobi), eigvecs in columns, sorted desc
__global__ void jacobi5(const float* __restrict__ Gin,
                        float* __restrict__ Vout, float* __restrict__ wout)
{
  float A[5][5], V[5][5];
  for (int i = 0; i < 5; ++i)
    for (int j = 0; j < 5; ++j) { A[i][j] = Gin[i * 5 + j]; V[i][j] = (i == j) ? 1.f : 0.f; }
  for (int sweep = 0; sweep < 30; ++sweep) {
    for (int p = 0; p < 4; ++p) {
      for (int q = p + 1; q < 5; ++q) {
        float apq = A[p][q];
        if (fabsf(apq) < 1e-12f) continue;
        float theta = (A[q][q] - A[p][p]) / (2.f * apq);
        float t = copysignf(1.f, theta) / (fabsf(theta) + sqrtf(theta * theta + 1.f));
        float c = 1.f / sqrtf(t * t + 1.f);
        float s = t * c;
        for (int k = 0; k < 5; ++k) {
          float akp = A[p][k], akq = A[q][k];
          A[p][k] = c * akp - s * akq;
          A[q][k] = s * akp + c * akq;
        }
        for (int k = 0; k < 5; ++k) {
          float akp = A[k][p], akq = A[k][q];
          A[k][p] = c * akp - s * akq;
          A[k][q] = s * akp + c * akq;
        }
        for (int k = 0; k < 5; ++k) {
          float vkp = V[k][p], vkq = V[k][q];
          V[k][p] = c * vkp - s * vkq;
          V[k][q] = s * vkp + c * vkq;
        }
      }
    }
  }
  float w[5];
  for (int i = 0; i < 5; ++i) w[i] = A[i][i];
  for (int i = 0; i < 4; ++i) {
    int best = i;
    for (int j = i + 1; j < 5; ++j) if (w[j] > w[best]) best = j;
    if (best != i) {
      float tw = w[i]; w[i] = w[best]; w[best] = tw;
      for (int k = 0; k < 5; ++k) { float tv = V[k][i]; V[k][i] = V[k][best]; V[k][best] = tv; }
    }
  }
  for (int i = 0; i < 5; ++i) wout[i] = w[i];
  for (int i = 0; i < 5; ++i)
    for (int j = 0; j < 5; ++j) Vout[i * 5 + j] = V[i][j];
}

// sigma = sqrt(max(w,0)): M5a = Us*diag(sqrt(sigma)), M5b = Us*diag(1/sqrt(sigma))
__global__ void make_scale5(const float* __restrict__ Us, const float* __restrict__ w,
                            float* __restrict__ M5a, float* __restrict__ M5b)
{
  float sq[5], isq[5];
  for (int j = 0; j < 5; ++j) {
    float sig = sqrtf(fmaxf(w[j], 0.f));
    float s = sqrtf(sig);
    sq[j] = s;
    isq[j] = (s > 1e-15f) ? 1.f / s : 0.f;
  }
  for (int i = 0; i < 5; ++i)
    for (int j = 0; j < 5; ++j) {
      M5a[i * 5 + j] = Us[i * 5 + j] * sq[j];
      M5b[i * 5 + j] = Us[i * 5 + j] * isq[j];
    }
}

// ||U V^T - U0 V0^T||_F from six 5x5 Grams (trace identities)
__global__ void finalize5(const float* __restrict__ GR, float* __restrict__ out)
{
  const float* Guu   = GR;
  const float* Gu0u  = GR + 25;
  const float* Gu0u0 = GR + 50;
  const float* Gvv   = GR + 75;
  const float* Gvv0  = GR + 100;
  const float* Gv0v0 = GR + 125;
  float t1 = 0.f, t2 = 0.f, t3 = 0.f;
  for (int a = 0; a < 5; ++a)
    for (int b = 0; b < 5; ++b) {
      t1 += Guu[a * 5 + b] * Gvv[b * 5 + a];
      t2 += Gu0u[a * 5 + b] * Gvv0[b * 5 + a];
      t3 += Gu0u0[a * 5 + b] * Gv0v0[b * 5 + a];
    }
  out[0] = sqrtf(fmaxf(t1 - 2.f * t2 + t3, 0.f));
}

// ---------------------------------------------------------------------------
extern "C" void kernel_launch(void* const* d_in, const int* in_sizes, int n_in,
                              void* d_out, int out_size, void* d_ws, size_t ws_size,
                              hipStream_t stream)
{
  (void)in_sizes; (void)n_in; (void)out_size; (void)ws_size;
  const float* Y0  = (const float*)d_in[0];
  const float* U0  = (const float*)d_in[1];
  const float* V0  = (const float*)d_in[2];
  const float* Om  = (const float*)d_in[3];
  const float* ths = (const float*)d_in[4];
  const float* stp = (const float*)d_in[5];
  float* out = (float*)d_out;
  float* ws  = (float*)d_ws;

  const int N = NDIM, M = NDIM, NUML = 15;
  const int P = N * 5;  // 30000 floats per panel

  float* P1 = ws + 0 * P;
  float* P2 = ws + 1 * P;
  float* Qn = ws + 2 * P;
  float* Qm = ws + 3 * P;
  float* Ua = ws + 4 * P;  float* Ub = ws + 5 * P;
  float* Va = ws + 6 * P;  float* Vb = ws + 7 * P;
  float* EV = ws + 8 * P;  float* ET = ws + 9 * P;
  float* Gg  = ws + 10 * P;          // small area
  float* Rin = Gg + 32;
  float* Vker = Gg + 64;
  float* Uker = Gg + 96;
  float* Us  = Gg + 128;
  float* wv  = Gg + 160;
  float* M5a = Gg + 192;
  float* M5b = Gg + 224;
  float* GR  = Gg + 256;             // 150
  float* PR  = Gg + 512;             // partR: CHUNKS * P
  float* PL  = PR + (size_t)CHUNKS * P;  // partL: STRIPS * P

  dim3 fg(CHUNKS, STRIPS);
  const int RB = (N + 127) / 128;

  auto Z = [&](float* p, int n) {
    zero_f32<<<(n + 255) / 256, 256, 0, stream>>>(p, n);
  };
  auto RED = [&](const float* part, float* o, int K) {
    reduce_sum<<<(P + 255) / 256, 256, 0, stream>>>(part, o, K, P);
  };

  // ---- layer 0: randomized SVD of A = clamp(Y0, th0) ----
  fused_pass<<<fg, TPB, 0, stream>>>(Y0, ths, 0, nullptr, Om, PR, nullptr, 0, N, M, TPC);
  RED(PR, P1, CHUNKS);
  Z(Gg, 25);
  gram5<<<64, 256, 0, stream>>>(P1, P1, Gg, N);
  chol_rinv5<<<1, 1, 0, stream>>>(Gg, Rin);
  rightmul5<<<RB, 128, 0, stream>>>(Qn, nullptr, P1, Rin, nullptr, 0, 0.f, N);

  for (int it = 0; it < 4; ++it) {
    fused_pass<<<fg, TPB, 0, stream>>>(Y0, ths, 0, Qn, nullptr, nullptr, PL, 0, N, M, TPC);
    RED(PL, P2, STRIPS);
    Z(Gg, 25);
    gram5<<<64, 256, 0, stream>>>(P2, P2, Gg, M);
    chol_rinv5<<<1, 1, 0, stream>>>(Gg, Rin);
    rightmul5<<<RB, 128, 0, stream>>>(Qm, nullptr, P2, Rin, nullptr, 0, 0.f, M);

    fused_pass<<<fg, TPB, 0, stream>>>(Y0, ths, 0, nullptr, Qm, PR, nullptr, 0, N, M, TPC);
    RED(PR, P1, CHUNKS);
    Z(Gg, 25);
    gram5<<<64, 256, 0, stream>>>(P1, P1, Gg, N);
    chol_rinv5<<<1, 1, 0, stream>>>(Gg, Rin);
    rightmul5<<<RB, 128, 0, stream>>>(Qn, nullptr, P1, Rin, nullptr, 0, 0.f, N);
  }

  // B^T = A^T Q (M x 5); B B^T via Gram
  fused_pass<<<fg, TPB, 0, stream>>>(Y0, ths, 0, Qn, nullptr, nullptr, PL, 0, N, M, TPC);
  RED(PL, P2, STRIPS);
  Z(Gg, 25);
  gram5<<<64, 256, 0, stream>>>(P2, P2, Gg, M);
  jacobi5<<<1, 1, 0, stream>>>(Gg, Us, wv);
  make_scale5<<<1, 1, 0, stream>>>(Us, wv, M5a, M5b);
  rightmul5<<<RB, 128, 0, stream>>>(Ua, nullptr, Qn, M5a, nullptr, 0, 0.f, N);
  rightmul5<<<RB, 128, 0, stream>>>(Va, nullptr, P2, M5b, nullptr, 0, 0.f, M);

  // ---- layers 1..14 ----
  float* Uc = Ua; float* Vc = Va; float* Un = Ub; float* Vn = Vb;
  for (int t = 1; t < NUML; ++t) {
    fused_pass<<<fg, TPB, 0, stream>>>(Y0, ths, t, Uc, Vc, PR, PL, 1, N, M, TPC);
    RED(PR, EV, CHUNKS);
    RED(PL, ET, STRIPS);

    Z(Gg, 25);
    gram5<<<64, 256, 0, stream>>>(Vc, Vc, Gg, M);
    inv5<<<1, 1, 0, stream>>>(Gg, Vker);
    Z(Gg, 25);
    gram5<<<64, 256, 0, stream>>>(Uc, Uc, Gg, N);
    inv5<<<1, 1, 0, stream>>>(Gg, Uker);

    rightmul5<<<RB, 128, 0, stream>>>(Un, Uc, EV, Vker, stp, t, 1.f, N);
    rightmul5<<<RB, 128, 0, stream>>>(Vn, Vc, ET, Uker, stp, t, 1.f, M);

    float* tu = Uc; Uc = Un; Un = tu;
    float* tv = Vc; Vc = Vn; Vn = tv;
  }

  // ---- final norm via 5x5 trace identities ----
  Z(GR, 150);
  gram5<<<64, 256, 0, stream>>>(Uc, Uc, GR + 0,   N);
  gram5<<<64, 256, 0, stream>>>(U0, Uc, GR + 25,  N);
  gram5<<<64, 256, 0, stream>>>(U0, U0, GR + 50,  N);
  gram5<<<64, 256, 0, stream>>>(Vc, Vc, GR + 75,  M);
  gram5<<<64, 256, 0, stream>>>(Vc, V0, GR + 100, M);
  gram5<<<64, 256, 0, stream>>>(V0, V0, GR + 125, M);
  finalize5<<<1, 1, 0, stream>>>(GR, out);
}